// HawkesProcess_6442450944054
// MI455X (gfx1250) — compile-verified
//
#include <hip/hip_runtime.h>

#define NT 10
#define BATCH 16
#define SEQ 512
#define STEPS 10
#define P2_SLICES 8
#define P2_TILES 33                    /* SEQ/16 + 1 rows of the (S+1) intervals */
#define P1_BLOCKS (BATCH * (SEQ / 16))
#define P2_GRID   (BATCH * P2_TILES)
#define P2_BLOCKS (P2_GRID * P2_SLICES)

typedef __attribute__((ext_vector_type(2))) float v2f;
typedef __attribute__((ext_vector_type(8))) float v8f;

// Tile row-reduction over j: D[m,n] = sum_k A[m,k]*1 + C[m,n].
// Exact f32 FMA accumulation; every column of D carries the row sums.
__device__ __forceinline__ v8f wmma_rowsum(v2f a, v8f c) {
  const v2f ones = {1.0f, 1.0f};
  return __builtin_amdgcn_wmma_f32_16x16x4_f32(false, a, false, ones,
                                               (short)0, c, false, false);
}

// ---------------------------------------------------------------------------
// Part 1: log-likelihood term.  One wave per (batch, 16-row i-tile).
// A element (m=i%16, k=j%4) lives at lane (m + 16*(k>=2)), VGPR k%2 per the
// CDNA5 16x4 f32 A layout. Branchless: unconditional b64 paired loads, masked
// dt (exp arg <= 0 always), cndmask-select into the A element.
// ---------------------------------------------------------------------------
__global__ __launch_bounds__(32) void hawkes_part1(
    const float* __restrict__ times, const int* __restrict__ types,
    const float* __restrict__ mu, const float* __restrict__ alpha,
    const float* __restrict__ beta, float* __restrict__ partials)
{
  __shared__ float sA[NT * NT], sB[NT * NT], sMu[NT];
  const int lane = threadIdx.x;
  for (int q = lane; q < NT * NT; q += 32) { sA[q] = alpha[q]; sB[q] = beta[q]; }
  if (lane < NT) sMu[lane] = mu[lane];
  __syncthreads();

  const int blk = blockIdx.x;
  const int b   = blk >> 5;            // 16 batches
  const int i0  = (blk & 31) << 4;     // 32 tiles of 16 rows
  const int m   = lane & 15;
  const int kb  = (lane >> 4) << 1;    // lanes 0-15 -> k {0,1}; 16-31 -> {2,3}
  const int i   = i0 + m;

  const float* __restrict__ tptr = times + b * SEQ;
  const int*   __restrict__ cptr = types + b * SEQ;
  const float ti = tptr[i];
  const int   ci = types[b * SEQ + i];

  v8f c = {};
  const int nch = (i0 + 16) >> 2;      // j-chunks of 4 covering j < i0+16
  for (int ch = 0; ch < nch; ++ch) {
    const int j0 = (ch << 2) + kb;     // even, 8B-aligned pair
    const float2 tj2 = *reinterpret_cast<const float2*>(tptr + j0);
    const int2   cj2 = *reinterpret_cast<const int2*>(cptr + j0);
    v2f a;
#pragma unroll
    for (int e = 0; e < 2; ++e) {
      const int   j   = j0 + e;
      const bool  val = (j < i);                       // strict causal mask
      const float tj  = e ? tj2.y : tj2.x;
      const int   cj  = e ? cj2.y : cj2.x;
      const float dt  = val ? (ti - tj) : 0.0f;        // keep exp arg <= 0
      const int   idx = cj * NT + ci;                  // alpha[t_j, t_i]
      const float g   = sA[idx] * __expf(-sB[idx] * dt);
      a[e] = val ? g : 0.0f;
    }
    c = wmma_rowsum(a, c);
  }

  // C/D layout: lane 0 holds (M=r, N=0) in c[r] for r=0..7; lane 16 holds M=8..15.
  float local = 0.0f;
  if (lane == 0 || lane == 16) {
    const int rb = (lane == 0) ? 0 : 8;
#pragma unroll
    for (int r = 0; r < 8; ++r) {
      const int ii = i0 + rb + r;
      local += c[r] + sMu[cptr[ii]];                   // + mu[event_types]
    }
  }
  const float other = __shfl(local, 16, 32);
  if (lane == 0) partials[blk] = local + other;
}

// ---------------------------------------------------------------------------
// Part 2: quadrature integral.  One 256-thread block (8 waves = 8 j-slices)
// per (batch, 16-interval s-tile).  Quadrature over the 11 points collapses
// analytically: sum_p e^{-beta(start + p*h - T_j)} = e^{-beta*d0} * sum_p r^p,
// r = e^{-beta*h} (Horner geometric sum) -- 5.5x fewer trans ops.  Branchless
// inner loop; j-reduction via WMMA f32 16x16x4 with B = ones.
// ---------------------------------------------------------------------------
__global__ __launch_bounds__(256) void hawkes_part2(
    const float* __restrict__ times, const int* __restrict__ types,
    const float* __restrict__ Tptr,
    const float* __restrict__ mu, const float* __restrict__ alpha,
    const float* __restrict__ beta, float* __restrict__ partials)
{
  __shared__ float sA[NT * NT], sB[NT * NT], sMu[NT];
  const int tid  = threadIdx.x;
  const int lane = tid & 31;
  const int wave = tid >> 5;           // wave == j-slice index
  for (int q = tid; q < NT * NT; q += 256) { sA[q] = alpha[q]; sB[q] = beta[q]; }
  if (tid < NT) sMu[tid] = mu[tid];
  __syncthreads();

  float musum = 0.0f;
#pragma unroll
  for (int k = 0; k < NT; ++k) musum += sMu[k];

  const int blk  = blockIdx.x;
  const int tile = blk % P2_TILES;
  const int b    = blk / P2_TILES;
  const int s0   = tile << 4;
  const int m    = lane & 15;
  const int kb   = (lane >> 4) << 1;
  const int s    = s0 + m;             // interval index, 0..SEQ (masked above)
  const float Tv = Tptr[0];
  const bool svalid = (s <= SEQ);

  const float* __restrict__ tptr = times + b * SEQ;
  const int*   __restrict__ cptr = types + b * SEQ;

  // Branchless start / h with clamped, always-in-range load indices.
  const int   sc    = (s > SEQ) ? SEQ : s;
  const float stm1  = tptr[(sc > 0 ? sc : 1) - 1];
  const float start = (sc == 0) ? 0.0f : stm1;
  const float tend  = tptr[(sc < SEQ) ? sc : 0];
  const float end   = (sc < SEQ) ? tend : Tv;
  const float h     = (end - start) * (1.0f / STEPS);  // >= 0, finite

  v8f c = {};
  const int jmax = (s0 + 16 < SEQ) ? (s0 + 16) : SEQ;
  const int nch  = jmax >> 2;
  for (int ch = wave; ch < nch; ch += P2_SLICES) {
    const int j0 = (ch << 2) + kb;
    const float2 tj2 = *reinterpret_cast<const float2*>(tptr + j0);
    const int2   cj2 = *reinterpret_cast<const int2*>(cptr + j0);
    v2f a;
#pragma unroll
    for (int e = 0; e < 2; ++e) {
      const int   j   = j0 + e;
      const bool  val = svalid && (j < s);             // hist mask: s > j
      const float tj  = e ? tj2.y : tj2.x;
      const int   cj  = e ? cj2.y : cj2.x;
      const float d0  = val ? (start - tj) : 0.0f;     // >= 0 either way
      const float* __restrict__ ar = &sA[cj * NT];
      const float* __restrict__ br = &sB[cj * NT];
      float g = 0.0f;
#pragma unroll
      for (int k = 0; k < NT; ++k) {
        const float bt = br[k];
        const float e0 = __expf(-bt * d0);
        const float r  = __expf(-bt * h);
        float pw = r, gs = 1.0f + r;                   // sum_{p=0..10} r^p
#pragma unroll
        for (int p = 2; p <= STEPS; ++p) { pw *= r; gs += pw; }
        g = fmaf(ar[k], e0 * gs, g);
      }
      a[e] = val ? g : 0.0f;
    }
    c = wmma_rowsum(a, c);
  }

  float local = 0.0f;
  if (lane == 0 || lane == 16) {
    const int rb = (lane == 0) ? 0 : 8;
#pragma unroll
    for (int r = 0; r < 8; ++r) {
      const int ss = s0 + rb + r;
      if (ss <= SEQ) {
        const float st = (ss == 0) ? 0.0f : tptr[ss - 1];
        const float en = (ss < SEQ) ? tptr[ss] : Tv;
        const float hh = (en - st) * (1.0f / STEPS);
        float term = c[r];
        if (wave == 0) term += (float)(STEPS + 1) * musum;  // count once
        local += hh * term;
      }
    }
  }
  const float other = __shfl(local, 16, 32);
  if (lane == 0) partials[blk * P2_SLICES + wave] = local + other;
}

// ---------------------------------------------------------------------------
// Final deterministic reduction: fixed-order serial sums of the partials.
// out = -(LL - integral) = integral - LL
// ---------------------------------------------------------------------------
__global__ void hawkes_final(const float* __restrict__ p1,
                             const float* __restrict__ p2,
                             float* __restrict__ out)
{
  if (threadIdx.x == 0 && blockIdx.x == 0) {
    float ll = 0.0f;
    for (int q = 0; q < P1_BLOCKS; ++q) ll += p1[q];
    float integ = 0.0f;
    for (int q = 0; q < P2_BLOCKS; ++q) integ += p2[q];
    out[0] = integ - ll;
  }
}

extern "C" void kernel_launch(void* const* d_in, const int* in_sizes, int n_in,
                              void* d_out, int out_size, void* d_ws, size_t ws_size,
                              hipStream_t stream) {
  (void)in_sizes; (void)n_in; (void)out_size; (void)ws_size;
  const float* times = (const float*)d_in[0];  // (16, 512) f32
  const int*   types = (const int*)d_in[1];    // (16, 512) i32
  const float* Tp    = (const float*)d_in[2];  // scalar f32
  const float* mu    = (const float*)d_in[3];  // (10,)
  const float* alpha = (const float*)d_in[4];  // (10, 10)
  const float* beta  = (const float*)d_in[5];  // (10, 10)

  float* p1 = (float*)d_ws;          // P1_BLOCKS partials
  float* p2 = p1 + P1_BLOCKS;        // P2_BLOCKS partials

  hawkes_part1<<<P1_BLOCKS, 32, 0, stream>>>(times, types, mu, alpha, beta, p1);
  hawkes_part2<<<P2_GRID, 256, 0, stream>>>(times, types, Tp, mu, alpha, beta, p2);
  hawkes_final<<<1, 1, 0, stream>>>(p1, p2, (float*)d_out);
}